// SequenceEmbedding_30923764532139
// MI455X (gfx1250) — compile-verified
//
#include <hip/hip_runtime.h>
#include <hip/hip_bf16.h>

// SequenceEmbedding: out[0][c][i][j] =
//   c < 4 : base_table[seq[i]][c]      (constant along j  -> scalar row fill)
//   c >= 4: base_table[seq[j]][c-4]    (constant along i  -> repeated row pattern)
//
// Pure streaming-store problem: 512 MiB out / 16 KB in. ~23 us floor at 23.3 TB/s.
// WMMA inapplicable (0 FLOPs/byte); the win is wide NT stores + async LDS->global DMA.

#define SEQ_L 4096
#define NB 4
constexpr int TPB = 256;   // 8 wave32s per block
constexpr int RPB = 16;    // rows per block -> grid.x = 256

typedef __attribute__((ext_vector_type(4))) float v4f;   // native vector for NT b128 stores

__global__ __launch_bounds__(TPB) void seq_embed_kernel(
    const int* __restrict__ seq,      // 4096 base ids (0..3)
    const float* __restrict__ table,  // 4x4 one-hot table
    float* __restrict__ out)          // 8 * 4096 * 4096 f32
{
    const int c    = blockIdx.z;            // channel 0..7
    const int row0 = blockIdx.x * RPB;      // first row of this block
    const int tid  = threadIdx.x;
    float* plane = out + (size_t)c * SEQ_L * SEQ_L;

    if (c < NB) {
        // ---- "row" half: each output row is a scalar fill of table[seq[i]][c] ----
        for (int r = 0; r < RPB; ++r) {
            const int i = row0 + r;
            const float v = table[seq[i] * NB + c];       // uniform -> scalarized
            const v4f vv = { v, v, v, v };
            v4f* rowp = reinterpret_cast<v4f*>(plane + (size_t)i * SEQ_L);
#pragma unroll
            for (int p = 0; p < 4; ++p) {
                // 512 MiB written once: non-temporal b128 stores, bypass cache retention
                __builtin_nontemporal_store(vv, rowp + tid + p * TPB);
            }
        }
    } else {
        // ---- "col" half: every row of this plane equals pattern[j] ----
        __shared__ float pat[SEQ_L];                      // 16 KB staged pattern
        const int cc = c - NB;

        __builtin_prefetch(seq + tid * (SEQ_L / TPB), 0, 1);  // global_prefetch_b8
        for (int j = tid; j < SEQ_L; j += TPB)
            pat[j] = table[seq[j] * NB + cc];
        __syncthreads();

        // LDS byte offset of pat[0]: low 32 bits of the generic shared pointer
        const unsigned lds_base = (unsigned)(uintptr_t)(&pat[0]);

        // Stream the same 16 KB LDS pattern out to every row via the async
        // LDS->global DMA path (tracked by ASYNCcnt, no VGPR data movement).
        for (int r = 0; r < RPB; ++r) {
            const unsigned rowbyte = (unsigned)(row0 + r) * (SEQ_L * 4u);
#pragma unroll
            for (int p = 0; p < 4; ++p) {
                const unsigned colb = (unsigned)tid * 16u + (unsigned)p * 4096u;
                const unsigned voff = rowbyte + colb;     // byte offset in plane (< 2^27)
                const unsigned lof  = lds_base + colb;    // per-lane LDS byte address
                asm volatile(
                    "global_store_async_from_lds_b128 %0, %1, %2 th:TH_STORE_NT"
                    :: "v"(voff), "v"(lof), "s"(plane)
                    : "memory");
            }
        }
        // Drain outstanding async stores (s_endpgm would also wait-idle, but be explicit)
        asm volatile("s_wait_asynccnt 0" ::: "memory");
    }
}

extern "C" void kernel_launch(void* const* d_in, const int* in_sizes, int n_in,
                              void* d_out, int out_size, void* d_ws, size_t ws_size,
                              hipStream_t stream) {
    const int*   seq   = (const int*)d_in[0];   // seq_ids (L,)
    const float* table = (const float*)d_in[1]; // base_table (4,4)
    float*       out   = (float*)d_out;         // (1, 8, L, L)
    (void)in_sizes; (void)n_in; (void)out_size; (void)d_ws; (void)ws_size;

    dim3 grid(SEQ_L / RPB, 1, 2 * NB);          // 256 row-chunks x 8 channels
    seq_embed_kernel<<<grid, TPB, 0, stream>>>(seq, table, out);
}